// ReprogrammingLayer_35983236006187
// MI455X (gfx1250) — compile-verified
//
#include <hip/hip_runtime.h>
#include <hip/hip_bf16.h>
#include <math.h>
#include <stdint.h>

typedef __attribute__((ext_vector_type(2))) float v2f;
typedef __attribute__((ext_vector_type(8))) float v8f;

#define D_LLM   768
#define N_HEADS 8
#define D_KEYS  128
#define DKH     1024   // D_KEYS * N_HEADS
#define S_SRC   1000
#define R_ROWS  12288  // B*N*T = 128*96
#define T_LEN   96
#define N_VARS  16

#define KC_GEMM 32     // K-chunk staged in LDS for the GEMM
#define KC_ATTN 40     // S-chunk staged in LDS for attention (25 * 40 = 1000)

// ---------------------------------------------------------------------------
// 1) Per-head column sums of Wk (768x1024 -> 768x8) and bk (1024 -> 8)
// ---------------------------------------------------------------------------
__global__ void reduce_wk_kernel(const float* __restrict__ Wk,
                                 const float* __restrict__ bk,
                                 float* __restrict__ WkS,   // [768*8], d*8+h
                                 float* __restrict__ bkS) { // [8]
    int idx = blockIdx.x * blockDim.x + threadIdx.x;
    if (idx < D_LLM * N_HEADS) {
        int d = idx >> 3, h = idx & 7;
        const float* p = Wk + (size_t)d * DKH + h * D_KEYS;
        float s = 0.f;
        for (int e = 0; e < D_KEYS; ++e) s += p[e];
        WkS[d * N_HEADS + h] = s;
    }
    if (idx < N_HEADS) {
        const float* p = bk + idx * D_KEYS;
        float s = 0.f;
        for (int e = 0; e < D_KEYS; ++e) s += p[e];
        bkS[idx] = s;
    }
}

// ---------------------------------------------------------------------------
// 2) Ksum[h,s] = SE[s,:] . WkS[:,h] + bkS[h]
// ---------------------------------------------------------------------------
__global__ void ksum_kernel(const float* __restrict__ SE,
                            const float* __restrict__ WkS,
                            const float* __restrict__ bkS,
                            float* __restrict__ Ksum) { // [8*S]
    int idx = blockIdx.x * blockDim.x + threadIdx.x;
    if (idx >= S_SRC * N_HEADS) return;
    int h = idx & 7, s = idx >> 3;
    const float* se = SE + (size_t)s * D_LLM;
    float acc = bkS[h];
    for (int d = 0; d < D_LLM; ++d) acc += se[d] * WkS[d * N_HEADS + h];
    Ksum[h * S_SRC + s] = acc;
}

// ---------------------------------------------------------------------------
// 3) per-head min / max of Ksum (closed-form softmax max for rank-1 logits)
// ---------------------------------------------------------------------------
__global__ void minmax_kernel(const float* __restrict__ Ksum,
                              float* __restrict__ maxK,
                              float* __restrict__ minK) {
    __shared__ float smax[256];
    __shared__ float smin[256];
    int h = blockIdx.x;
    float mx = -INFINITY, mn = INFINITY;
    for (int s = threadIdx.x; s < S_SRC; s += blockDim.x) {
        float v = Ksum[h * S_SRC + s];
        mx = fmaxf(mx, v);
        mn = fminf(mn, v);
    }
    smax[threadIdx.x] = mx;
    smin[threadIdx.x] = mn;
    __syncthreads();
    for (int off = 128; off > 0; off >>= 1) {
        if (threadIdx.x < (unsigned)off) {
            smax[threadIdx.x] = fmaxf(smax[threadIdx.x], smax[threadIdx.x + off]);
            smin[threadIdx.x] = fminf(smin[threadIdx.x], smin[threadIdx.x + off]);
        }
        __syncthreads();
    }
    if (threadIdx.x == 0) { maxK[h] = smax[0]; minK[h] = smin[0]; }
}

// ---------------------------------------------------------------------------
// 4) fp32 WMMA GEMM with bias and async-LDS B staging:
//    C[M,N] = A[M,K] @ B[K,N] + bias[N]
//    block = 256 threads (8 waves); block tile 128 x 64; wave -> 16 x 64
//    grid  = (N/64, ceil(M/128)); K, N multiples of KC_GEMM / 64.
// ---------------------------------------------------------------------------
__global__ __launch_bounds__(256)
void gemm_f32_wmma_kernel(const float* __restrict__ A,
                          const float* __restrict__ B,
                          const float* __restrict__ bias,
                          float* __restrict__ C,
                          int M, int N, int K) {
    __shared__ float ldsB[2][KC_GEMM * 64];

    const int tid   = threadIdx.x;
    const int lane  = tid & 31;
    const int wave  = tid >> 5;
    const int row0  = blockIdx.y * 128 + wave * 16;
    const int col0  = blockIdx.x * 64;
    const int rlo   = lane & 15;
    const int khalf = (lane >> 4) * 2;

    // clamp instead of predicate: keeps EXEC all-ones through the WMMA loop
    const int   rA     = row0 + rlo;
    const int   rcl    = (rA < M) ? rA : (M - 1);
    const float avalid = (rA < M) ? 1.f : 0.f;
    const float* Arow  = A + (size_t)rcl * K;

    // async stage of B[k .. k+KC-1][col0 .. col0+63] into ldsB[buf]
    auto stage = [&](int k, int buf) {
        unsigned base = (unsigned)(uintptr_t)(&ldsB[buf][0]);
#pragma unroll
        for (int i = 0; i < 2; ++i) {
            int f4 = tid + i * 256;          // float4 index in the 32x64 tile
            int kk = f4 >> 4;                // 16 float4 per row
            int c4 = f4 & 15;
            const float* g = B + (size_t)(k + kk) * N + (col0 + c4 * 4);
            unsigned l = base + (unsigned)(f4 * 16);
            asm volatile("global_load_async_to_lds_b128 %0, %1, off"
                         :: "v"(l), "v"(g) : "memory");
        }
    };

    v8f acc[4] = {v8f{}, v8f{}, v8f{}, v8f{}};

    stage(0, 0);
    const int nch = K / KC_GEMM;
    for (int c = 0; c < nch; ++c) {
        if (c + 1 < nch) {
            stage((c + 1) * KC_GEMM, (c + 1) & 1);
            asm volatile("s_wait_asynccnt 0x2" ::: "memory"); // prev chunk landed
        } else {
            asm volatile("s_wait_asynccnt 0x0" ::: "memory");
        }
        __syncthreads();

        const float* bb = &ldsB[c & 1][0];
        const int kbase = c * KC_GEMM;
#pragma unroll
        for (int kk = 0; kk < KC_GEMM; kk += 4) {
            const float* ap = Arow + kbase + kk + khalf;
            v2f a;
            a.x = ap[0] * avalid;
            a.y = ap[1] * avalid;
#pragma unroll
            for (int ct = 0; ct < 4; ++ct) {
                const float* bp = bb + (kk + khalf) * 64 + ct * 16 + rlo;
                v2f bf;
                bf.x = bp[0];
                bf.y = bp[64];
                acc[ct] = __builtin_amdgcn_wmma_f32_16x16x4_f32(
                    false, a, false, bf, (short)0, acc[ct], false, false);
            }
        }
        __syncthreads();
    }

    // D layout: lane l, vgpr v -> row = v + 8*(l>>4), col = l&15
#pragma unroll
    for (int v = 0; v < 8; ++v) {
        int rr = row0 + v + 8 * (lane >> 4);
        if (rr >= M) continue;
#pragma unroll
        for (int ct = 0; ct < 4; ++ct) {
            int cc = col0 + ct * 16 + rlo;
            C[(size_t)rr * N + cc] = acc[ct][v] + bias[cc];
        }
    }
}

// ---------------------------------------------------------------------------
// 5) Fused softmax(q * Ksum) @ V_h with async-LDS V staging.
//    block = 256 threads (8 waves); block tile = 128 rows x 1 head
//    grid = (R/128, H)
//    attn[r, h*128 + e] = sum_s softmax_s(scale*q[r]*Ksum[h,s]) * V[s, h*128+e]
// ---------------------------------------------------------------------------
__global__ __launch_bounds__(256)
void attn_kernel(const float* __restrict__ ts,    // [8,96,16]
                 const float* __restrict__ Ksum,  // [8,S]
                 const float* __restrict__ Vbuf,  // [S, 1024]
                 const float* __restrict__ maxK,
                 const float* __restrict__ minK,
                 float* __restrict__ attn) {      // [R, 1024]
    __shared__ float ldsV[2][KC_ATTN * 128];
    __shared__ float denom_lds[8][16];

    const int tid   = threadIdx.x;
    const int lane  = tid & 31;
    const int wave  = tid >> 5;
    const int h     = blockIdx.y;
    const int row0  = blockIdx.x * 128 + wave * 16;
    const int rlo   = lane & 15;
    const int khalf = (lane >> 4) * 2;
    const float scale = 0.08838834764831845f;     // 1/sqrt(128)

    // q for fragment row: r = (b*16+n)*96 + t ; ts is [b][t][n]
    int r  = row0 + rlo;
    int bn = r / T_LEN;
    int t  = r - bn * T_LEN;
    int b  = bn >> 4;
    int n  = bn & 15;
    float q = ts[(b * T_LEN + t) * N_VARS + n];
    if (q != q) q = 0.f;                          // nan -> 0
    float m  = scale * q * (q >= 0.f ? maxK[h] : minK[h]);
    float qs = scale * q;

    const float* Kh = Ksum + h * S_SRC;
    const float* Vh = Vbuf + h * D_KEYS;

    // async stage of V[s0 .. s0+KC_ATTN-1][h*128 .. +127] into ldsV[buf]
    auto stage = [&](int s0, int buf) {
        unsigned base = (unsigned)(uintptr_t)(&ldsV[buf][0]);
#pragma unroll
        for (int i = 0; i < 5; ++i) {
            int f4 = tid + i * 256;          // float4 index in the 40x128 tile
            int kk = f4 >> 5;                // 32 float4 per row
            int c4 = f4 & 31;
            const float* g = Vh + (size_t)(s0 + kk) * DKH + c4 * 4;
            unsigned l = base + (unsigned)(f4 * 16);
            asm volatile("global_load_async_to_lds_b128 %0, %1, off"
                         :: "v"(l), "v"(g) : "memory");
        }
    };

    v8f acc[8] = {v8f{}, v8f{}, v8f{}, v8f{}, v8f{}, v8f{}, v8f{}, v8f{}};
    float dpart = 0.f;

    stage(0, 0);
    const int nch = S_SRC / KC_ATTN;          // 25
    for (int c = 0; c < nch; ++c) {
        if (c + 1 < nch) {
            stage((c + 1) * KC_ATTN, (c + 1) & 1);
            asm volatile("s_wait_asynccnt 0x5" ::: "memory");
        } else {
            asm volatile("s_wait_asynccnt 0x0" ::: "memory");
        }
        __syncthreads();

        const float* vb = &ldsV[c & 1][0];
        const int s0 = c * KC_ATTN;
#pragma unroll
        for (int kk = 0; kk < KC_ATTN; kk += 4) {
            float k0 = Kh[s0 + kk + khalf];
            float k1 = Kh[s0 + kk + khalf + 1];
            v2f a;
            a.x = __expf(qs * k0 - m);
            a.y = __expf(qs * k1 - m);
            dpart += a.x + a.y;
#pragma unroll
            for (int ct = 0; ct < 8; ++ct) {
                const float* vp = vb + (kk + khalf) * 128 + ct * 16 + rlo;
                v2f bf;
                bf.x = vp[0];
                bf.y = vp[128];
                acc[ct] = __builtin_amdgcn_wmma_f32_16x16x4_f32(
                    false, a, false, bf, (short)0, acc[ct], false, false);
            }
        }
        __syncthreads();
    }

    // full-row denominator: lanes l and l+16 hold the two K-halves of row rlo
    dpart += __shfl_xor(dpart, 16, 32);
    if (lane < 16) denom_lds[wave][lane] = dpart;
    __syncthreads();

#pragma unroll
    for (int v = 0; v < 8; ++v) {
        int rr    = row0 + v + 8 * (lane >> 4);
        float inv = 1.0f / denom_lds[wave][v + 8 * (lane >> 4)];
#pragma unroll
        for (int ct = 0; ct < 8; ++ct) {
            int cc = h * D_KEYS + ct * 16 + rlo;
            attn[(size_t)rr * DKH + cc] = acc[ct][v] * inv;
        }
    }
}

// ---------------------------------------------------------------------------
// launch
// ---------------------------------------------------------------------------
extern "C" void kernel_launch(void* const* d_in, const int* in_sizes, int n_in,
                              void* d_out, int out_size, void* d_ws, size_t ws_size,
                              hipStream_t stream) {
    const float* ts  = (const float*)d_in[0]; // time_series      [8,96,16]
    const float* SE  = (const float*)d_in[1]; // source_embedding [1000,768]
    const float* VE  = (const float*)d_in[2]; // value_embedding  [1000,768]
    const float* Wk  = (const float*)d_in[3]; // [768,1024]
    const float* bk  = (const float*)d_in[4]; // [1024]
    const float* Wv  = (const float*)d_in[5]; // [768,1024]
    const float* bv  = (const float*)d_in[6]; // [1024]
    const float* Wo  = (const float*)d_in[7]; // [1024,768]
    const float* bo  = (const float*)d_in[8]; // [768]
    float*       out = (float*)d_out;         // [12288,768]

    float* ws   = (float*)d_ws;
    float* Vb   = ws;                    // 1,024,000 floats  (1000x1024)
    float* Ks   = ws + 1024000;          // 8,000             (8x1000)
    float* WkS  = ws + 1032000;          // 6,144             (768x8)
    float* bkS  = ws + 1038144;          // 8
    float* mxK  = ws + 1038152;          // 8
    float* mnK  = ws + 1038160;          // 8
    float* attn = ws + 1040000;          // 12,582,912        (12288x1024)

    // 1) per-head reductions of Wk / bk
    reduce_wk_kernel<<<(D_LLM * N_HEADS + 255) / 256, 256, 0, stream>>>(Wk, bk, WkS, bkS);

    // 2) Ksum[h,s]
    ksum_kernel<<<(S_SRC * N_HEADS + 255) / 256, 256, 0, stream>>>(SE, WkS, bkS, Ks);

    // 3) per-head min/max
    minmax_kernel<<<N_HEADS, 256, 0, stream>>>(Ks, mxK, mnK);

    // 4) V = VE @ Wv + bv   (M=1000, N=1024, K=768)
    {
        dim3 grid(DKH / 64, (S_SRC + 127) / 128);
        gemm_f32_wmma_kernel<<<grid, 256, 0, stream>>>(VE, Wv, bv, Vb, S_SRC, DKH, D_LLM);
    }

    // 5) fused softmax + P@V   (grid: 96 row-blocks x 8 heads, 8 waves each)
    {
        dim3 grid(R_ROWS / 128, N_HEADS);
        attn_kernel<<<grid, 256, 0, stream>>>(ts, Ks, Vb, mxK, mnK, attn);
    }

    // 6) out = attn @ Wo + bo  (M=12288, N=768, K=1024)
    {
        dim3 grid(D_LLM / 64, R_ROWS / 128);
        gemm_f32_wmma_kernel<<<grid, 256, 0, stream>>>(attn, Wo, bo, out, R_ROWS, D_LLM, DKH);
    }
}